// MultiVariateHawkesProcess_67954972557801
// MI455X (gfx1250) — compile-verified
//
#include <hip/hip_runtime.h>
#include <math.h>

#define D10 10
#define LOG2E 1.4426950408889634f

typedef __attribute__((ext_vector_type(2))) float f32x2;
typedef __attribute__((ext_vector_type(8))) float v8f;

// ---------------- CDNA5 async global->LDS staging (probe-gated) -------------
// Probe result: builtin exists and takes generic `int*` params.
#if __has_builtin(__builtin_amdgcn_global_load_async_to_lds_b32)
#define HAVE_ASYNC_LDS 1
#define ASYNC_G2L_B32(gsrc, ldst)                                              \
  __builtin_amdgcn_global_load_async_to_lds_b32((int*)(gsrc), (int*)(ldst), 0, \
                                                0)
static __device__ __forceinline__ void async_wait0() {
#if __has_builtin(__builtin_amdgcn_s_wait_asynccnt)
  __builtin_amdgcn_s_wait_asynccnt(0);
#else
  asm volatile("s_wait_asynccnt 0x0" ::: "memory");
#endif
}
#else
#define HAVE_ASYNC_LDS 0
#endif

static __device__ __forceinline__ float softplusf(float x) {
  // matches jax.nn.softplus = log1p(exp(x)); stable for large x
  return (x > 20.0f) ? x : log1pf(expf(x));
}

// ---------------------------------------------------------------------------
// Kernel A: pairwise interaction row-sums (strict lower triangle).
// Block = 256 threads = 256 rows; blockIdx.y selects a CH-wide j chunk.
// part[by*N + i] = sum_{j in chunk, j < i} a*b*exp(-b*(t_i - t_j))
// ---------------------------------------------------------------------------
template <int CH>
__global__ __launch_bounds__(256) void hawkes_pairwise(
    const float* __restrict__ tp, const int* __restrict__ et,
    const float* __restrict__ la, const float* __restrict__ lb,
    float* __restrict__ part, int N) {
  __shared__ f32x2 s_tab[D10 * D10];  // {alpha*beta, beta*log2(e)}
  __shared__ float s_t[CH];
  __shared__ int s_e[CH];

  const int tid = threadIdx.x;
  const int Rs = blockIdx.x * 256;
  const int Js = blockIdx.y * CH;
  const int row = Rs + tid;

  if (tid < D10 * D10) {
    float a = softplusf(la[tid]);
    float b = softplusf(lb[tid]);
    f32x2 v;
    v.x = a * b;
    v.y = b * LOG2E;
    s_tab[tid] = v;
  }

  // Entire chunk is at j >= i for every row of this block: own the partial
  // slot with an explicit zero (ws is poisoned) and quit. Uniform branch.
  if (Js >= Rs + 256) {
    part[blockIdx.y * N + row] = 0.0f;
    return;
  }

  // Stage this chunk's t_j / e_j into LDS (async DMA path on CDNA5).
  constexpr int PT = CH / 256;
#pragma unroll
  for (int k = 0; k < PT; ++k) {
    int idx = tid + k * 256;
#if HAVE_ASYNC_LDS
    ASYNC_G2L_B32(tp + (Js + idx), &s_t[idx]);
    ASYNC_G2L_B32(et + (Js + idx), &s_e[idx]);
#else
    s_t[idx] = tp[Js + idx];
    s_e[idx] = et[Js + idx];
#endif
  }
#if HAVE_ASYNC_LDS
  async_wait0();
#endif
  __syncthreads();

  const float ti = tp[row];
  const int ei = et[row];
  const f32x2* trow = &s_tab[ei * D10];

  float acc = 0.0f;
  const bool full = (Js + CH <= Rs);  // every j in chunk is < every row
  if (full) {
#pragma unroll 4
    for (int jj = 0; jj < CH; ++jj) {
      float tj = s_t[jj];   // uniform broadcast read
      int ej = s_e[jj];     // uniform broadcast read
      f32x2 ab = trow[ej];  // conflict-free gather (stride-10 table)
      acc += ab.x * __builtin_amdgcn_exp2f(ab.y * (tj - ti));
    }
  } else {
    const int jmax = row - Js;  // jj < jmax  <=>  j < i  (>= 0 here)
#pragma unroll 4
    for (int jj = 0; jj < CH; ++jj) {
      float tj = s_t[jj];
      int ej = s_e[jj];
      f32x2 ab = trow[ej];
      float v = ab.x * __builtin_amdgcn_exp2f(ab.y * (tj - ti));
      acc += (jj < jmax) ? v : 0.0f;
    }
  }
  part[blockIdx.y * N + row] = acc;
}

// ---------------------------------------------------------------------------
// Kernel B: intensities -> sum(log), integral term, final scalar.
// Final 256-way reduction via V_WMMA_F32_16X16X4_F32 against a ones matrix.
// ---------------------------------------------------------------------------
__global__ __launch_bounds__(256) void hawkes_finish(
    const float* __restrict__ tp, const int* __restrict__ et,
    const float* __restrict__ mu, const float* __restrict__ la,
    const float* __restrict__ lb, const int* __restrict__ Traw,
    const float* __restrict__ part, int N, int NJ, float* __restrict__ out) {
  __shared__ float s_a[D10 * D10];
  __shared__ float s_b2[D10 * D10];
  __shared__ float s_mu[D10];
  __shared__ float s_part[256];

  const int tid = threadIdx.x;
  if (tid < D10 * D10) {
    s_a[tid] = softplusf(la[tid]);
    s_b2[tid] = softplusf(lb[tid]) * LOG2E;
  }
  if (tid < D10) s_mu[tid] = softplusf(mu[tid]);
  __syncthreads();

  // T arrives as a 1-element array; dtype (int vs float bits) is ambiguous.
  int bits = Traw[0];
  float asf = __int_as_float(bits);
  float T = (asf > 1e-6f && asf < 1e12f) ? asf : (float)bits;

  float local = 0.0f;
  for (int i = tid; i < N; i += 256) {
    float ti = tp[i];
    int ei = et[i];
    float inten = s_mu[ei];
    for (int c = 0; c < NJ; ++c) inten += part[c * N + i];
    local += logf(inten);
    float dT = T - ti;
#pragma unroll
    for (int d = 0; d < D10; ++d) {
      local -= s_a[d * D10 + ei] *
               (1.0f - __builtin_amdgcn_exp2f(-s_b2[d * D10 + ei] * dT));
    }
  }
  if (tid == 0) {
    float msum = 0.0f;
#pragma unroll
    for (int d = 0; d < D10; ++d) msum += s_mu[d];
    local -= T * msum;
  }
  s_part[tid] = local;
  __syncthreads();

#if __has_builtin(__builtin_amdgcn_wmma_f32_16x16x4_f32)
  // View s_part as M[16][16]; C += A_k(16x4) * ones(4x16) accumulates row
  // sums into every column of C. Column 0 lives in lanes 0 and 16.
  if (tid < 32) {  // exactly wave 0 under wave32; EXEC all-ones
    const int m = tid & 15;
    const int half = tid >> 4;
    v8f c = {};
    f32x2 ones;
    ones.x = 1.0f;
    ones.y = 1.0f;
#pragma unroll
    for (int k = 0; k < 4; ++k) {
      f32x2 a;  // ISA 16x4 f32 A layout: lanes0-15 hold K={0,1}, 16-31 K={2,3}
      a.x = s_part[m * 16 + k * 4 + (half ? 2 : 0)];
      a.y = s_part[m * 16 + k * 4 + (half ? 3 : 1)];
      c = __builtin_amdgcn_wmma_f32_16x16x4_f32(false, a, false, ones,
                                                (short)0, c, false, false);
    }
    float s = c[0] + c[1] + c[2] + c[3] + c[4] + c[5] + c[6] + c[7];
    float tot = __shfl(s, 0, 32) + __shfl(s, 16, 32);  // M=0..7 + M=8..15
    if (tid == 0) out[0] = tot;
  }
#else
  for (int stride = 128; stride > 0; stride >>= 1) {
    if (tid < stride) s_part[tid] += s_part[tid + stride];
    __syncthreads();
  }
  if (tid == 0) out[0] = s_part[0];
#endif
}

// ---------------------------------------------------------------------------
extern "C" void kernel_launch(void* const* d_in, const int* in_sizes, int n_in,
                              void* d_out, int out_size, void* d_ws,
                              size_t ws_size, hipStream_t stream) {
  const float* tp = (const float*)d_in[0];  // time_points (N,)
  const int* et = (const int*)d_in[1];      // event_types (N,) int32
  const float* mu = (const float*)d_in[2];  // mu (D,)
  const float* la = (const float*)d_in[3];  // log_alpha (D,D)
  const float* lb = (const float*)d_in[4];  // log_beta  (D,D)
  const int* Tr = (const int*)d_in[5];      // T scalar (1,)
  float* out = (float*)d_out;
  float* part = (float*)d_ws;

  const int N = in_sizes[0];  // 8192

  int NJ = 8;
  if ((size_t)NJ * (size_t)N * sizeof(float) > ws_size) NJ = 1;

  if (NJ == 8) {
    dim3 grid(N / 256, 8);
    hawkes_pairwise<1024><<<grid, 256, 0, stream>>>(tp, et, la, lb, part, N);
  } else {
    dim3 grid(N / 256, 1);
    hawkes_pairwise<8192><<<grid, 256, 0, stream>>>(tp, et, la, lb, part, N);
  }
  hawkes_finish<<<1, 256, 0, stream>>>(tp, et, mu, la, lb, Tr, part, N, NJ,
                                       out);
}